// GCN_3513283248845
// MI455X (gfx1250) — compile-verified
//
#include <hip/hip_runtime.h>

#define NN 50000
#define NE 800000
#define NLAB 50
#define EMB 64
#define HID 96

typedef __attribute__((ext_vector_type(2))) float v2f;
typedef __attribute__((ext_vector_type(8))) float v8f;

// ---------------- degree / norm kernels ----------------

__global__ void zero2_kernel(float* __restrict__ a, float* __restrict__ b, int n) {
    int i = blockIdx.x * blockDim.x + threadIdx.x;
    if (i < n) { a[i] = 0.0f; b[i] = 0.0f; }
}

__global__ void edge_deg_kernel(const int* __restrict__ src, const int* __restrict__ dst,
                                float* __restrict__ odeg, float* __restrict__ ideg, int e) {
    int i = blockIdx.x * blockDim.x + threadIdx.x;
    if (i < e) {
        atomicAdd(&odeg[src[i]], 1.0f);
        atomicAdd(&ideg[dst[i]], 1.0f);
    }
}

// in-place: deg -> rsqrt(max(deg,1))
__global__ void norm_kernel(float* __restrict__ od, float* __restrict__ id, int n) {
    int i = blockIdx.x * blockDim.x + threadIdx.x;
    if (i < n) {
        od[i] = rsqrtf(fmaxf(od[i], 1.0f));
        id[i] = rsqrtf(fmaxf(id[i], 1.0f));
    }
}

// ---------------- feature prep kernels ----------------

// h_scaled[i][f] = emb[label[i]][f] * out_norm[i]   (block = EMB threads, grid = NN)
__global__ void embed_scale_kernel(const int* __restrict__ labels, const float* __restrict__ emb,
                                   const float* __restrict__ onorm, float* __restrict__ out) {
    int i = blockIdx.x;
    int f = threadIdx.x;
    out[(long)i * EMB + f] = emb[(long)labels[i] * EMB + f] * onorm[i];
}

// dst[i][f] = src[i][f] * norm[i]   (block = d threads, grid = NN)
__global__ void scale_rows_kernel(const float* __restrict__ h, const float* __restrict__ norm,
                                  float* __restrict__ out, int d) {
    int i = blockIdx.x;
    int f = threadIdx.x;
    out[(long)i * d + f] = h[(long)i * d + f] * norm[i];
}

__global__ void zero_buf_kernel(float* __restrict__ p, long n) {
    long i = (long)blockIdx.x * blockDim.x + threadIdx.x;
    if (i < n) p[i] = 0.0f;
}

// agg[dst[e]][f] += hs[src[e]][f]   (block = d threads, grid = NE)
__global__ void scatter_add_kernel(const float* __restrict__ hs, float* __restrict__ agg,
                                   const int* __restrict__ src, const int* __restrict__ dst, int d) {
    int e = blockIdx.x;
    int f = threadIdx.x;
    int s = src[e], t = dst[e];
    atomicAdd(&agg[(long)t * d + f], hs[(long)s * d + f]);
}

// ---------------- WMMA GEMM: out = relu((A * rowScale[row]) @ W + b) ----------------
// One wave32 per 16x16 output tile, V_WMMA_F32_16X16X4_F32 over K.
// W (K x Nout) + bias staged into LDS, zero-padded to Npad columns.

__global__ __launch_bounds__(256) void gcn_gemm_wmma(
    const float* __restrict__ A, const float* __restrict__ rowScale,
    const float* __restrict__ W, const float* __restrict__ bias,
    float* __restrict__ out, int K, int Nout, int Npad,
    int tilesN, int totalTiles, int relu)
{
    __shared__ float smem[HID * HID + HID];   // max 96*96 W + 96 bias = 37248 B
    const int tid = threadIdx.x;
    const int KN = K * Npad;

    for (int idx = tid; idx < KN; idx += 256) {
        int k = idx / Npad, n = idx % Npad;
        smem[idx] = (n < Nout) ? W[k * Nout + n] : 0.0f;
    }
    for (int idx = tid; idx < Npad; idx += 256)
        smem[KN + idx] = (idx < Nout) ? bias[idx] : 0.0f;
    __syncthreads();

    const int wave = tid >> 5;
    const int lane = tid & 31;
    const int tileId = blockIdx.x * 8 + wave;
    if (tileId >= totalTiles) return;

    const int m0 = (tileId / tilesN) * 16;
    const int n0 = (tileId % tilesN) * 16;

    // ISA 16x4 f32 A layout: lanes 0-15 -> M=lane, K=k0+{0,1}; lanes 16-31 -> K=k0+{2,3}
    const int rowA = m0 + (lane & 15);
    const int koff = (lane >> 4) * 2;
    const int colB = lane & 15;

    // C layout: VGPR r: lanes 0-15 -> M=r, lanes 16-31 -> M=8+r; N = lane%16
    v8f c;
    const float bval = smem[KN + n0 + colB];
#pragma unroll
    for (int r = 0; r < 8; ++r) c[r] = bval;

    const float s = rowScale[rowA];                 // fused in_norm scaling
    const float* arow = A + (long)rowA * K;
    const float* bcol = smem + n0 + colB;

    for (int k0 = 0; k0 < K; k0 += 4) {
        v2f a, b;
        a.x = arow[k0 + koff] * s;
        a.y = arow[k0 + koff + 1] * s;
        b.x = bcol[(k0 + koff) * Npad];
        b.y = bcol[(k0 + koff + 1) * Npad];
        c = __builtin_amdgcn_wmma_f32_16x16x4_f32(
                /*neg_a=*/false, a, /*neg_b=*/false, b,
                /*c_mod=*/(short)0, c, /*reuse_a=*/false, /*reuse_b=*/false);
    }

    const int rbase = m0 + (lane >> 4) * 8;
    const int col = n0 + colB;
    if (col < Nout) {
#pragma unroll
        for (int r = 0; r < 8; ++r) {
            float v = c[r];
            if (relu) v = fmaxf(v, 0.0f);
            out[(long)(rbase + r) * Nout + col] = v;
        }
    }
}

// ---------------- driver ----------------

extern "C" void kernel_launch(void* const* d_in, const int* in_sizes, int n_in,
                              void* d_out, int out_size, void* d_ws, size_t ws_size,
                              hipStream_t stream) {
    const int*   labels = (const int*)d_in[0];
    const int*   src    = (const int*)d_in[1];
    const int*   dst    = (const int*)d_in[2];
    const float* emb    = (const float*)d_in[3];
    const float* W1     = (const float*)d_in[4];
    const float* b1     = (const float*)d_in[5];
    const float* W2     = (const float*)d_in[6];
    const float* b2     = (const float*)d_in[7];
    const float* W3     = (const float*)d_in[8];
    const float* b3     = (const float*)d_in[9];
    float* out = (float*)d_out;

    float* w = (float*)d_ws;
    float* onorm = w;                    // NN   (deg -> norm in place)
    float* inorm = w + NN;               // NN
    float* bufA  = w + 2 * (size_t)NN;   // NN*HID : gemm output h
    float* bufB  = bufA + (size_t)NN * HID; // NN*HID : h * out_norm
    float* bufC  = bufB + (size_t)NN * HID; // NN*HID : aggregated features

    const int TPB = 256;

    // degrees -> norms
    zero2_kernel<<<(NN + TPB - 1) / TPB, TPB, 0, stream>>>(onorm, inorm, NN);
    edge_deg_kernel<<<(NE + TPB - 1) / TPB, TPB, 0, stream>>>(src, dst, onorm, inorm, NE);
    norm_kernel<<<(NN + TPB - 1) / TPB, TPB, 0, stream>>>(onorm, inorm, NN);

    // ---- layer 1: [NN,EMB] -> [NN,HID], relu ----
    embed_scale_kernel<<<NN, EMB, 0, stream>>>(labels, emb, onorm, bufB);
    zero_buf_kernel<<<(int)(((long)NN * EMB + TPB - 1) / TPB), TPB, 0, stream>>>(bufC, (long)NN * EMB);
    scatter_add_kernel<<<NE, EMB, 0, stream>>>(bufB, bufC, src, dst, EMB);
    {
        int tilesN = HID / 16;                       // 6
        int total  = (NN / 16) * tilesN;             // 18750
        gcn_gemm_wmma<<<(total + 7) / 8, 256, 0, stream>>>(
            bufC, inorm, W1, b1, bufA, EMB, HID, HID, tilesN, total, 1);
    }

    // ---- layer 2: [NN,HID] -> [NN,HID], relu ----
    scale_rows_kernel<<<NN, HID, 0, stream>>>(bufA, onorm, bufB, HID);
    zero_buf_kernel<<<(int)(((long)NN * HID + TPB - 1) / TPB), TPB, 0, stream>>>(bufC, (long)NN * HID);
    scatter_add_kernel<<<NE, HID, 0, stream>>>(bufB, bufC, src, dst, HID);
    {
        int tilesN = HID / 16;
        int total  = (NN / 16) * tilesN;
        gcn_gemm_wmma<<<(total + 7) / 8, 256, 0, stream>>>(
            bufC, inorm, W2, b2, bufA, HID, HID, HID, tilesN, total, 1);
    }

    // ---- layer 3: [NN,HID] -> [NN,NLAB], no relu ----
    scale_rows_kernel<<<NN, HID, 0, stream>>>(bufA, onorm, bufB, HID);
    zero_buf_kernel<<<(int)(((long)NN * HID + TPB - 1) / TPB), TPB, 0, stream>>>(bufC, (long)NN * HID);
    scatter_add_kernel<<<NE, HID, 0, stream>>>(bufB, bufC, src, dst, HID);
    {
        int Npad   = 64;                             // pad 50 -> 64
        int tilesN = Npad / 16;                      // 4
        int total  = (NN / 16) * tilesN;             // 12500
        gcn_gemm_wmma<<<(total + 7) / 8, 256, 0, stream>>>(
            bufC, inorm, W3, b3, out, HID, NLAB, Npad, tilesN, total, 0);
    }
}